// LanguageDetectionModel_25159918420248
// MI455X (gfx1250) — compile-verified
//
#include <hip/hip_runtime.h>
#include <hip/hip_bf16.h>

#define VOCAB  128256
#define HID    256
#define NLANG  100
#define BATCH  64
#define SEQ    2048

#define NT     7        // N tiles (112 padded langs / 16)
#define LPAD   112
#define KPAIRS (HID/2)  // 128 K-pairs

typedef __attribute__((ext_vector_type(2))) float v2f;
typedef __attribute__((ext_vector_type(8))) float v8f;

// Exact float max via sign-aware integer atomics (works vs -inf init, no NaNs here).
__device__ __forceinline__ void atomicMaxF32(float* addr, float v) {
    if (v >= 0.0f) atomicMax((int*)addr, __float_as_int(v));
    else           atomicMin((unsigned int*)addr, (unsigned int)__float_as_int(v));
}

__global__ void init_out_kernel(float* out, int n) {
    int i = blockIdx.x * blockDim.x + threadIdx.x;
    if (i < n) out[i] = -__builtin_inff();
}

__global__ __launch_bounds__(128) void langdet_wmma_kernel(
    const int*   __restrict__ token_ids,     // (B,S)
    const float* __restrict__ embeddings,    // (V,H)
    const float* __restrict__ token_weights, // (V,1)
    const float* __restrict__ proj_w,        // (L,H)
    const float* __restrict__ proj_b,        // (L)
    float*       __restrict__ out)           // (B,L)
{
    // B matrix staged K-pair-major, pre-swizzled for ds_load_b64 fragments:
    // b_lds[p*LPAD + n] = { proj_w[n][2p], proj_w[n][2p+1] }   (0 for n >= NLANG)
    __shared__ float2 b_lds[KPAIRS * LPAD];   // 114688 bytes

    const int tidx = threadIdx.x;
    for (int idx = tidx; idx < KPAIRS * LPAD; idx += 128) {
        int p = idx / LPAD;
        int n = idx - p * LPAD;
        float2 v = make_float2(0.0f, 0.0f);
        if (n < NLANG) {
            v.x = proj_w[n * HID + 2 * p];
            v.y = proj_w[n * HID + 2 * p + 1];
        }
        b_lds[idx] = v;
    }
    __syncthreads();

    const int lane = tidx & 31;
    const int wave = tidx >> 5;
    const int b    = blockIdx.x;           // batch row
    const int seg  = blockIdx.y;           // 256-token segment
    const int lrow = lane & 15;            // token-in-tile (A) / lang-in-tile (B,C)
    const int half = lane >> 4;            // K sub-phase select

    const int tokseg = b * SEQ + seg * 256;

    // This wave owns M-tiles wave*4 .. wave*4+3 (64 tokens). Gather setup + prefetch.
    int         ids [4];
    float       wts [4];
    const float* rows[4];
#pragma unroll
    for (int j = 0; j < 4; ++j) {
        int tok  = tokseg + (wave * 4 + j) * 16 + lrow;
        ids[j]   = token_ids[tok];
        wts[j]   = token_weights[ids[j]];
        rows[j]  = embeddings + (size_t)ids[j] * HID;
#pragma unroll
        for (int i = 0; i < 4; ++i)   // each half-wave covers 512B of the 1KB row
            __builtin_prefetch(rows[j] + half * 128 + i * 32, 0, 1);
    }

    float runmax[NT];
#pragma unroll
    for (int t = 0; t < NT; ++t) runmax[t] = -__builtin_inff();

    for (int pair = 0; pair < 2; ++pair) {
        const int j0 = pair * 2;
        const float* row0 = rows[j0 + 0];
        const float* row1 = rows[j0 + 1];
        const float  w0   = wts[j0 + 0];
        const float  w1   = wts[j0 + 1];

        v8f acc0[NT], acc1[NT];
#pragma unroll
        for (int t = 0; t < NT; ++t) { acc0[t] = (v8f)0.0f; acc1[t] = (v8f)0.0f; }

        for (int kk = 0; kk < HID; kk += 4) {
            const int ko = kk + 2 * half;                  // this lane's K pair
            float2 a0 = *(const float2*)(row0 + ko);
            float2 a1 = *(const float2*)(row1 + ko);
            v2f av0 = { a0.x * w0, a0.y * w0 };
            v2f av1 = { a1.x * w1, a1.y * w1 };
            const float2* bp = &b_lds[((kk >> 1) + half) * LPAD + lrow];
#pragma unroll
            for (int t = 0; t < NT; ++t) {
                float2 bb = bp[t * 16];
                v2f bv = { bb.x, bb.y };
                acc0[t] = __builtin_amdgcn_wmma_f32_16x16x4_f32(
                    false, av0, false, bv, (short)0, acc0[t], false, false);
                acc1[t] = __builtin_amdgcn_wmma_f32_16x16x4_f32(
                    false, av1, false, bv, (short)0, acc1[t], false, false);
            }
        }

        // Fold 16 token rows of both C tiles into per-lane running max.
#pragma unroll
        for (int t = 0; t < NT; ++t) {
            float m = fmaxf(acc0[t][0], acc1[t][0]);
#pragma unroll
            for (int r = 1; r < 8; ++r) m = fmaxf(m, fmaxf(acc0[t][r], acc1[t][r]));
            runmax[t] = fmaxf(runmax[t], m);
        }
    }

    // Merge the two half-wave token groups, add bias, publish via atomic max.
#pragma unroll
    for (int t = 0; t < NT; ++t) {
        float m = fmaxf(runmax[t], __shfl_xor(runmax[t], 16, 32));
        int lang = t * 16 + lrow;
        if (lane < 16 && lang < NLANG)
            atomicMaxF32(&out[b * NLANG + lang], m + proj_b[lang]);
    }
}

extern "C" void kernel_launch(void* const* d_in, const int* in_sizes, int n_in,
                              void* d_out, int out_size, void* d_ws, size_t ws_size,
                              hipStream_t stream) {
    const int*   token_ids     = (const int*)  d_in[0];
    const float* embeddings    = (const float*)d_in[1];
    const float* token_weights = (const float*)d_in[2];
    const float* proj_w        = (const float*)d_in[3];
    const float* proj_b        = (const float*)d_in[4];
    float* out = (float*)d_out;

    const int n_out = BATCH * NLANG;
    hipLaunchKernelGGL(init_out_kernel, dim3((n_out + 255) / 256), dim3(256), 0, stream,
                       out, n_out);

    dim3 grid(BATCH, SEQ / 256);   // 64 x 8 blocks, 4 waves each
    hipLaunchKernelGGL(langdet_wmma_kernel, grid, dim3(128), 0, stream,
                       token_ids, embeddings, token_weights, proj_w, proj_b, out);
}